// TrainableTFNLayer_50010599195150
// MI455X (gfx1250) — compile-verified
//
#include <hip/hip_runtime.h>

#define B_ 8
#define N_ 4096
#define D_ 128
#define M_ 1024
#define KC 64             // K-chunk (rows) staged in LDS per TDM transfer
#define RS 144            // padded LDS row stride in floats (128 data + 16 TDM pad)
#define CH (KC * RS)      // floats per LDS chunk buffer
#define LN_EPSF 1e-5f
#define LOG2E_F 1.4426950408889634f

typedef __attribute__((ext_vector_type(2))) float    v2f;
typedef __attribute__((ext_vector_type(8))) float    v8f;
typedef __attribute__((ext_vector_type(4))) unsigned v4u;
typedef __attribute__((ext_vector_type(8))) unsigned v8u;

__device__ __forceinline__ float clampf(float x, float lo, float hi) {
    return fminf(fmaxf(x, lo), hi);
}

// Low 32 bits of a flat pointer to LDS == wave-relative LDS byte offset
// (aperture lives in the high 32 bits; ISA 10.2: LDS address is the low 32 bits).
__device__ __forceinline__ unsigned lds_off_u32(const void* p) {
    return (unsigned)(unsigned long long)(uintptr_t)p;
}

// ---------------------------------------------------------------------------
// Issue one TDM 2D tile load: KC rows x 128 f32 from row-major global tensor
// into LDS with 16-DWORD padding after every 128 DWORDs (row stride -> 144).
// D# group0: count=1 | lds_addr | global_addr[56:0] | type=2.
// D# group1: data_size=4B, pad_enable, pad_interval=128dw(code 6),
//            pad_amount=16dw(code 15), dims/tile/stride for 2D.
// ---------------------------------------------------------------------------
__device__ __forceinline__ void tdm_load_tile(unsigned lds_byte_off,
                                              const float* gsrc,
                                              unsigned tensor_dim1) {
    unsigned long long ga = (unsigned long long)(uintptr_t)gsrc;
    v4u g0 = {
        1u,                                                   // count=1, user D#
        lds_byte_off,                                         // lds_addr
        (unsigned)ga,                                         // global_addr[31:0]
        (unsigned)((ga >> 32) & 0x1FFFFFFull) | (2u << 30)    // addr[56:32] | type=2
    };
    v8u g1 = {
        (2u << 16) | (1u << 20) | (6u << 22) | (15u << 25),   // dsize=4B|pad|int=128|amt=16
        ((unsigned)D_ & 0xffffu) << 16,                       // tensor_dim0 lo16 @ [63:48]
        (tensor_dim1 & 0xffffu) << 16,                        // dim0 hi16=0 | dim1 lo16
        (tensor_dim1 >> 16) | ((unsigned)D_ << 16),           // dim1 hi16 | tile_dim0=128
        (unsigned)KC,                                         // tile_dim1=KC | tile_dim2=0
        (unsigned)D_,                                         // tensor_dim0_stride=128
        0u,                                                   // stride0 hi | stride1 lo (2D: unused)
        0u
    };
    asm volatile("tensor_load_to_lds %0, %1" :: "s"(g0), "s"(g1) : "memory");
}

// ---------------------------------------------------------------------------
// Kernel 1: field[b,m,d] = sum_n exp(-(g_m - p_bn)^2 / 2s^2) * emb[b,n,d]
// Per-batch GEMM (1024 x 4096 x 128) on V_WMMA_F32_16X16X4_F32, TDM-fed,
// double-buffered LDS. 128 blocks; 8 waves; wave tile = 16 m x 64 d.
// ---------------------------------------------------------------------------
__global__ __launch_bounds__(256)
void tfn_project(const float* __restrict__ emb, const float* __restrict__ pos,
                 const float* __restrict__ sigma_p, float* __restrict__ field)
{
    __shared__ float Xs[2][CH];     // double-buffered padded chunk [k][d(+pad)]
    __shared__ float ps[2][KC];     // double-buffered positions chunk

    const int b      = blockIdx.x >> 4;              // 16 m-blocks per batch
    const int mblock = (blockIdx.x & 15) * 64;
    const int tid    = threadIdx.x;
    const int wave   = tid >> 5;
    const int lane   = tid & 31;
    const int ln15   = lane & 15;
    const int khalf  = (lane < 16) ? 0 : 2;          // ISA A/B K-split per half-wave

    const float s  = clampf(sigma_p[0], 0.01f, 2.0f);
    const float c2 = -LOG2E_F / (2.0f * s * s);      // exp(x) = exp2(x*log2e)

    const int   mtile = mblock + (wave >> 1) * 16;
    const int   dbase = (wave & 1) * 64;
    const float g     = (float)(mtile + ln15) * (1.0f / (float)(M_ - 1));

    const float* embB = emb + (size_t)b * N_ * D_;
    const float* posB = pos + (size_t)b * N_;

    v8f acc[4] = {};
    const int NCH = N_ / KC;

    // prologue: prefetch chunk 0
    if (wave == 0) {
        tdm_load_tile(lds_off_u32(&Xs[0][0]), embB, (unsigned)N_);
        ps[0][lane]      = posB[lane];
        ps[0][lane + 32] = posB[lane + 32];
    }

    for (int c = 0; c < NCH; ++c) {
        const int cur = c & 1;
        if (wave == 0) {
            if (c + 1 < NCH) {
                const int kcn = (c + 1) * KC;
                tdm_load_tile(lds_off_u32(&Xs[cur ^ 1][0]), embB + (size_t)kcn * D_,
                              (unsigned)N_);
                ps[cur ^ 1][lane]      = posB[kcn + lane];
                ps[cur ^ 1][lane + 32] = posB[kcn + lane + 32];
                __builtin_amdgcn_s_wait_tensorcnt(1);   // chunk c complete (in-order)
            } else {
                __builtin_amdgcn_s_wait_tensorcnt(0);
            }
        }
        __syncthreads();                                // chunk c visible to all waves

        const float* Xc  = &Xs[cur][0];
        const float* psc = &ps[cur][0];
        #pragma unroll 4
        for (int ks = 0; ks < KC; ks += 4) {
            float p0  = psc[ks + khalf];
            float p1  = psc[ks + khalf + 1];
            float dx0 = g - p0, dx1 = g - p1;
            v2f a;
            a.x = __builtin_amdgcn_exp2f(dx0 * dx0 * c2);
            a.y = __builtin_amdgcn_exp2f(dx1 * dx1 * c2);
            #pragma unroll
            for (int j = 0; j < 4; ++j) {
                const int dj = dbase + j * 16 + ln15;
                v2f bf;                                  // conflict-free: +288dw = +32 banks
                bf.x = Xc[(ks + khalf) * RS + dj];
                bf.y = Xc[(ks + khalf + 1) * RS + dj];
                acc[j] = __builtin_amdgcn_wmma_f32_16x16x4_f32(
                    false, a, false, bf, (short)0, acc[j], false, false);
            }
        }
        __syncthreads();                                // all waves done with buf 'cur'
    }

    // C/D layout: VGPR r -> row mtile + r (lanes 0-15) / mtile + 8 + r (lanes 16-31)
    float* fB = field + (size_t)b * M_ * D_;
    const int mbase = mtile + ((lane < 16) ? 0 : 8);
    #pragma unroll
    for (int j = 0; j < 4; ++j)
        #pragma unroll
        for (int r = 0; r < 8; ++r)
            fB[(size_t)(mbase + r) * D_ + dbase + j * 16 + ln15] = acc[j][r];
}

// ---------------------------------------------------------------------------
// Kernel 2: one explicit diffusion step (interior Laplacian along m).
// ---------------------------------------------------------------------------
__global__ __launch_bounds__(256)
void tfn_diffuse(const float* __restrict__ src, float* __restrict__ dst,
                 const float* __restrict__ alpha_p, const float* __restrict__ dt_p)
{
    const float a  = clampf(alpha_p[0], 0.01f, 1.0f);
    const float dt = clampf(dt_p[0], 0.001f, 0.1f);
    const float c  = a * dt;
    int idx = blockIdx.x * 256 + threadIdx.x;
    if (idx >= B_ * M_ * D_) return;
    int m = (idx >> 7) & (M_ - 1);            // layout [b][m][d], D=128, M=1024
    float f   = src[idx];
    float out = f;
    if (m >= 1 && m <= M_ - 2)
        out = f + c * (src[idx + D_] - 2.0f * f + src[idx - D_]);
    dst[idx] = out;
}

// ---------------------------------------------------------------------------
// Kernel 3: sampled[b,n,d] = sum_m K[b,n,m] * field[b,m,d], then fused
// residual + LN1 + LN2. Per-batch GEMM (4096 x 1024 x 128), TDM-fed.
// 256 blocks; 8 waves; wave tile = 16 n-rows x 128 d-cols.
// ---------------------------------------------------------------------------
__global__ __launch_bounds__(256)
void tfn_sample_ln(const float* __restrict__ emb, const float* __restrict__ pos,
                   const float* __restrict__ field,
                   const float* __restrict__ sigma_p,
                   const float* __restrict__ w1, const float* __restrict__ b1,
                   const float* __restrict__ w2, const float* __restrict__ b2,
                   float* __restrict__ out)
{
    __shared__ float Fs[2][CH];

    const int b      = blockIdx.x >> 5;              // 32 n-blocks per batch
    const int nblock = (blockIdx.x & 31) * 128;
    const int tid    = threadIdx.x;
    const int wave   = tid >> 5;
    const int lane   = tid & 31;
    const int ln15   = lane & 15;
    const int khalf  = (lane < 16) ? 0 : 2;

    const float s    = clampf(sigma_p[0], 0.01f, 2.0f);
    const float c2   = -LOG2E_F / (2.0f * s * s);
    const float ginv = 1.0f / (float)(M_ - 1);

    const int   ntile = nblock + wave * 16;
    const float p     = pos[(size_t)b * N_ + ntile + ln15];  // lane's token position

    const float* fB = field + (size_t)b * M_ * D_;

    v8f acc[8] = {};
    const int NCH = M_ / KC;

    if (wave == 0)
        tdm_load_tile(lds_off_u32(&Fs[0][0]), fB, (unsigned)M_);

    for (int c = 0; c < NCH; ++c) {
        const int cur = c & 1;
        if (wave == 0) {
            if (c + 1 < NCH) {
                tdm_load_tile(lds_off_u32(&Fs[cur ^ 1][0]),
                              fB + (size_t)(c + 1) * KC * D_, (unsigned)M_);
                __builtin_amdgcn_s_wait_tensorcnt(1);
            } else {
                __builtin_amdgcn_s_wait_tensorcnt(0);
            }
        }
        __syncthreads();

        const float* Fc = &Fs[cur][0];
        const int kc = c * KC;
        #pragma unroll 4
        for (int ks = 0; ks < KC; ks += 4) {
            float g0  = (float)(kc + ks + khalf) * ginv;   // grid coord is analytic
            float g1  = g0 + ginv;
            float dx0 = g0 - p, dx1 = g1 - p;
            v2f a;
            a.x = __builtin_amdgcn_exp2f(dx0 * dx0 * c2);
            a.y = __builtin_amdgcn_exp2f(dx1 * dx1 * c2);
            #pragma unroll
            for (int j = 0; j < 8; ++j) {
                const int dj = j * 16 + ln15;
                v2f bf;
                bf.x = Fc[(ks + khalf) * RS + dj];
                bf.y = Fc[(ks + khalf + 1) * RS + dj];
                acc[j] = __builtin_amdgcn_wmma_f32_16x16x4_f32(
                    false, a, false, bf, (short)0, acc[j], false, false);
            }
        }
        __syncthreads();
    }

    // ----- fused epilogue: residual + LayerNorm x2 over D=128 -----
    float w1v[8], b1v[8], w2v[8], b2v[8];
    #pragma unroll
    for (int j = 0; j < 8; ++j) {
        int dj = j * 16 + ln15;
        w1v[j] = w1[dj]; b1v[j] = b1[dj];
        w2v[j] = w2[dj]; b2v[j] = b2[dj];
    }

    const int nbase = ntile + ((lane < 16) ? 0 : 8);
    #pragma unroll
    for (int r = 0; r < 8; ++r) {
        const int n = nbase + r;
        const float* erow = emb + ((size_t)b * N_ + n) * D_;

        float v[8]; float sum = 0.f;
        #pragma unroll
        for (int j = 0; j < 8; ++j) { v[j] = acc[j][r] + erow[j * 16 + ln15]; sum += v[j]; }
        #pragma unroll
        for (int msk = 1; msk <= 8; msk <<= 1) sum += __shfl_xor(sum, msk, 32);
        float mu = sum * (1.0f / 128.0f);

        float vs = 0.f;
        #pragma unroll
        for (int j = 0; j < 8; ++j) { float t = v[j] - mu; vs += t * t; }
        #pragma unroll
        for (int msk = 1; msk <= 8; msk <<= 1) vs += __shfl_xor(vs, msk, 32);
        float rstd = __builtin_amdgcn_rsqf(vs * (1.0f / 128.0f) + LN_EPSF);

        float y[8]; float sum2 = 0.f;
        #pragma unroll
        for (int j = 0; j < 8; ++j) { y[j] = (v[j] - mu) * rstd * w1v[j] + b1v[j]; sum2 += y[j]; }
        #pragma unroll
        for (int msk = 1; msk <= 8; msk <<= 1) sum2 += __shfl_xor(sum2, msk, 32);
        float mu2 = sum2 * (1.0f / 128.0f);

        float vs2 = 0.f;
        #pragma unroll
        for (int j = 0; j < 8; ++j) { float t = y[j] - mu2; vs2 += t * t; }
        #pragma unroll
        for (int msk = 1; msk <= 8; msk <<= 1) vs2 += __shfl_xor(vs2, msk, 32);
        float rstd2 = __builtin_amdgcn_rsqf(vs2 * (1.0f / 128.0f) + LN_EPSF);

        float* orow = out + ((size_t)b * N_ + n) * D_;
        #pragma unroll
        for (int j = 0; j < 8; ++j)
            orow[j * 16 + ln15] = (y[j] - mu2) * rstd2 * w2v[j] + b2v[j];
    }
}

// ---------------------------------------------------------------------------
extern "C" void kernel_launch(void* const* d_in, const int* in_sizes, int n_in,
                              void* d_out, int out_size, void* d_ws, size_t ws_size,
                              hipStream_t stream) {
    const float* emb   = (const float*)d_in[0];
    const float* pos   = (const float*)d_in[1];
    const float* sigma = (const float*)d_in[2];
    const float* alpha = (const float*)d_in[3];
    const float* dt    = (const float*)d_in[4];
    const float* w1    = (const float*)d_in[5];
    const float* b1    = (const float*)d_in[6];
    const float* w2    = (const float*)d_in[7];
    const float* b2    = (const float*)d_in[8];
    float* outp = (float*)d_out;

    float* field0 = (float*)d_ws;
    float* field1 = field0 + (size_t)B_ * M_ * D_;

    // 1) project tokens onto grid (TDM-fed WMMA GEMM)
    tfn_project<<<B_ * (M_ / 64), 256, 0, stream>>>(emb, pos, sigma, field0);

    // 2) three diffusion steps, ping-pong in workspace (final result in field1)
    const int nElem = B_ * M_ * D_;
    const int nb = (nElem + 255) / 256;
    tfn_diffuse<<<nb, 256, 0, stream>>>(field0, field1, alpha, dt);
    tfn_diffuse<<<nb, 256, 0, stream>>>(field1, field0, alpha, dt);
    tfn_diffuse<<<nb, 256, 0, stream>>>(field0, field1, alpha, dt);

    // 3) sample back at token positions + residual + double LayerNorm
    tfn_sample_ln<<<B_ * (N_ / 128), 256, 0, stream>>>(
        emb, pos, field1, sigma, w1, b1, w2, b2, outp);
}